// GCNLayer_27934467293298
// MI455X (gfx1250) — compile-verified
//
#include <hip/hip_runtime.h>

typedef float v2f __attribute__((ext_vector_type(2)));
typedef float v8f __attribute__((ext_vector_type(8)));

// ---------------------------------------------------------------------------
// Kernel 1: zero the output, init degree to 1 (self loop contribution)
// ---------------------------------------------------------------------------
__global__ void __launch_bounds__(256)
gcn_init_kernel(float* __restrict__ out, unsigned* __restrict__ deg,
                int out_n, int n) {
    int i = blockIdx.x * blockDim.x + threadIdx.x;
    if (i < out_n) out[i] = 0.0f;
    if (i < n) deg[i] = 1u;
}

// ---------------------------------------------------------------------------
// Kernel 2: degree accumulation via integer atomics (src and dst direction)
// ---------------------------------------------------------------------------
__global__ void __launch_bounds__(256)
gcn_deg_kernel(const int* __restrict__ src, const int* __restrict__ dst,
               unsigned* __restrict__ deg, int E) {
    int i = blockIdx.x * blockDim.x + threadIdx.x;
    if (i < E) {
        atomicAdd(&deg[src[i]], 1u);
        atomicAdd(&deg[dst[i]], 1u);
    }
}

// ---------------------------------------------------------------------------
// Kernel 3: deg (u32) -> dinv = rsqrt(deg) (f32), in place
// ---------------------------------------------------------------------------
__global__ void __launch_bounds__(256)
gcn_rsqrt_kernel(void* __restrict__ degbuf, int n) {
    int i = blockIdx.x * blockDim.x + threadIdx.x;
    if (i < n) {
        unsigned d = ((const unsigned*)degbuf)[i];
        float df = (float)(d < 1u ? 1u : d);
        ((float*)degbuf)[i] = rsqrtf(df);
    }
}

// ---------------------------------------------------------------------------
// Kernel 4: h = x @ W^T + b via V_WMMA_F32_16X16X4_F32.
// One wave computes one 16x16 tile of h. K = 64 -> 16 WMMA steps.
//
// A (16x4 f32) lane layout (ISA 7.12.2): lanes 0-15 hold row M=lane with
// VGPR0=K0, VGPR1=K1; lanes 16-31 hold row M=lane-16 with VGPR0=K2, VGPR1=K3.
// B (4x16) mirrors it; since B = W^T, B[k][n] = W[n][k], so the per-lane
// addressing of W rows is identical to the A addressing of x rows:
// both are contiguous float2 loads at [(base + (lane&15))*64 + k + 2*(lane>>4)].
// EXEC must stay all-ones for WMMA: tile guard is wave-uniform.
// ---------------------------------------------------------------------------
__global__ void __launch_bounds__(256)
gcn_gemm_wmma_kernel(const float* __restrict__ x,
                     const float* __restrict__ W,
                     const float* __restrict__ bias,
                     float* __restrict__ h,
                     int totalTiles) {
    int wave = threadIdx.x >> 5;
    int lane = threadIdx.x & 31;
    int tile = blockIdx.x * (blockDim.x >> 5) + wave;
    if (tile >= totalTiles) return;               // wave-uniform exit

    int row0 = (tile >> 2) << 4;                  // 16-row tile of x / h
    int col0 = (tile & 3) << 4;                   // 16-col tile of D_OUT=64
    int half = lane >> 4;                         // K sub-group (0 or 1)
    int lr   = lane & 15;

    const float* arow = x + (size_t)(row0 + lr) * 64 + 2 * half;
    const float* brow = W + (size_t)(col0 + lr) * 64 + 2 * half;

    v8f acc = {};
#pragma unroll
    for (int k = 0; k < 64; k += 4) {
        v2f a = *(const v2f*)(arow + k);
        v2f b = *(const v2f*)(brow + k);
        // 8 args: (neg_a, A, neg_b, B, c_mod, C, reuse_a, reuse_b)
        acc = __builtin_amdgcn_wmma_f32_16x16x4_f32(
            false, a, false, b, (short)0, acc, false, false);
    }

    // D layout: VGPR v -> element (M = v + 8*half, N = lr)
    float bv = bias[col0 + lr];
    float* hout = h + (size_t)(row0 + 8 * half) * 64 + col0 + lr;
#pragma unroll
    for (int v = 0; v < 8; ++v) hout[(size_t)v * 64] = acc[v] + bv;
}

// ---------------------------------------------------------------------------
// Kernel 5: message scatter. One wave per augmented edge; each lane handles
// features {2*lane, 2*lane+1} -> coalesced 256B gather of h[s], then two
// hardware fp32 atomics (global_atomic_add_f32) into out[d].
// out (25.6MB) is L2-resident (192MB L2) so atomics resolve in-cache.
// Augmented edge list: [src->dst | dst->src | self loops], E' = 2E + N.
// ---------------------------------------------------------------------------
__global__ void __launch_bounds__(256)
gcn_scatter_kernel(const float* __restrict__ h,
                   const float* __restrict__ dinv,
                   const int* __restrict__ src,
                   const int* __restrict__ dst,
                   float* __restrict__ out,
                   int E, int Eaug) {
    long long gid = (long long)blockIdx.x * blockDim.x + threadIdx.x;
    int e    = (int)(gid >> 5);
    int lane = (int)(gid & 31);
    if (e >= Eaug) return;

    int s, d;
    if (e < E)            { s = src[e];      d = dst[e];      }
    else if (e < 2 * E)   { s = dst[e - E];  d = src[e - E];  }
    else                  { s = e - 2 * E;   d = s;           }

    float nrm = dinv[s] * dinv[d];
    v2f hv = *(const v2f*)(h + (size_t)s * 64 + 2 * lane);
    float* op = out + (size_t)d * 64 + 2 * lane;
    // Guaranteed hardware GLOBAL_ATOMIC_ADD_F32 (no CAS-loop fallback).
    unsafeAtomicAdd(op,     hv.x * nrm);
    unsafeAtomicAdd(op + 1, hv.y * nrm);
}

// ---------------------------------------------------------------------------
// Launch
// ---------------------------------------------------------------------------
extern "C" void kernel_launch(void* const* d_in, const int* in_sizes, int n_in,
                              void* d_out, int out_size, void* d_ws, size_t ws_size,
                              hipStream_t stream) {
    const float* x  = (const float*)d_in[0];
    const int*   ei = (const int*)d_in[1];     // [2, E] int32
    const float* W  = (const float*)d_in[2];   // [64, 64]
    const float* b  = (const float*)d_in[3];   // [64]

    const int N = in_sizes[0] / 64;
    const int E = in_sizes[1] / 2;
    const int Eaug = 2 * E + N;
    const int* src = ei;
    const int* dst = ei + E;

    float* out = (float*)d_out;

    // Workspace layout: [deg/dinv: N*4 bytes][pad to 256][h: N*64*4 bytes]
    char* ws = (char*)d_ws;
    unsigned* deg  = (unsigned*)ws;
    float*    dinv = (float*)ws;
    size_t hoff = ((size_t)N * 4 + 255) & ~(size_t)255;
    float* h = (float*)(ws + hoff);

    const int BLK = 256;

    // 1) init out + deg
    {
        int n = out_size > N ? out_size : N;
        gcn_init_kernel<<<(n + BLK - 1) / BLK, BLK, 0, stream>>>(out, deg, out_size, N);
    }
    // 2) degree accumulation
    gcn_deg_kernel<<<(E + BLK - 1) / BLK, BLK, 0, stream>>>(src, dst, deg, E);
    // 3) dinv = rsqrt(deg)
    gcn_rsqrt_kernel<<<(N + BLK - 1) / BLK, BLK, 0, stream>>>((void*)deg, N);
    // 4) h = x @ W^T + b   (N is a multiple of 16 for this problem: 100000)
    {
        int rowTiles = N / 16;
        int totalTiles = rowTiles * 4;           // D_OUT=64 -> 4 col tiles
        int wavesPerBlock = BLK / 32;
        int blocks = (totalTiles + wavesPerBlock - 1) / wavesPerBlock;
        gcn_gemm_wmma_kernel<<<blocks, BLK, 0, stream>>>(x, W, b, h, totalTiles);
    }
    // 5) scatter-add messages
    {
        long long threads = (long long)Eaug * 32;
        long long blocks = (threads + BLK - 1) / BLK;
        gcn_scatter_kernel<<<(int)blocks, BLK, 0, stream>>>(h, dinv, src, dst, out, E, Eaug);
    }
    (void)n_in; (void)ws_size;
}